// VectorQuantizer_60035052863654
// MI455X (gfx1250) — compile-verified
//
#include <hip/hip_runtime.h>
#include <stdint.h>

// Problem constants (match the reference): B=32, N=4096 (=64x64), K=1024, D=256.
#define TOK_N   4096
#define EMB_D   256
#define TILE    64          // tokens per workgroup
#define PITCH   260         // f32 per LDS row: 256 + 4 pad (keeps 16B align, 2-way max conflict)
#define THREADS 256

typedef int v4i __attribute__((ext_vector_type(4)));
typedef __attribute__((address_space(1))) v4i* gv4i_ptr;   // global AS pointer to int4
typedef __attribute__((address_space(3))) v4i* lv4i_ptr;   // LDS AS pointer to int4

__device__ __forceinline__ void async_load_b128(void* lds_dst, void* gsrc) {
#if __has_builtin(__builtin_amdgcn_global_load_async_to_lds_b128)
    __builtin_amdgcn_global_load_async_to_lds_b128(
        (gv4i_ptr)gsrc, (lv4i_ptr)lds_dst, /*imm offset*/0, /*cpol*/0);
#else
    uint32_t lds_off = (uint32_t)(uintptr_t)(__attribute__((address_space(3))) void*)lds_dst;
    asm volatile("global_load_async_to_lds_b128 %0, %1, off"
                 :: "v"(lds_off), "v"(gsrc)
                 : "memory");
#endif
}

__device__ __forceinline__ void wait_async0() {
#if __has_builtin(__builtin_amdgcn_s_wait_asynccnt)
    __builtin_amdgcn_s_wait_asynccnt(0);
#else
    asm volatile("s_wait_asynccnt 0" ::: "memory");
#endif
}

__global__ __launch_bounds__(THREADS)
void vq_embed_gather_tr_kernel(const int* __restrict__ idx,
                               const float* __restrict__ emb,
                               float* __restrict__ out) {
    __shared__ float s_tile[TILE * PITCH];  // 66,560 B staged tile
    __shared__ int   s_idx[TILE];

    const int tid  = threadIdx.x;
    const int tok0 = blockIdx.x * TILE;          // first flat token of this tile
    const int b    = tok0 / TOK_N;               // batch index
    const int n0   = tok0 - b * TOK_N;           // spatial offset within batch

    if (tid < TILE) s_idx[tid] = idx[tok0 + tid];
    __syncthreads();

    // ---- Phase 1: async-gather 64 codebook rows (1 KB each) into LDS ----
    // 4096 x 16B chunks, 16 per thread; each 64-thread group streams one row
    // contiguously (512 B per wave per instruction) straight from L2 to LDS,
    // no VGPR round-trip, tracked on ASYNCcnt. Codebook loads stay RT (hot).
#pragma unroll
    for (int i = 0; i < 16; ++i) {
        const int chunk = i * THREADS + tid;     // 0..4095
        const int r     = chunk >> 6;            // local token 0..63
        const int c16   = chunk & 63;            // 16B chunk within the row
        const float* g  = emb + (size_t)s_idx[r] * EMB_D + c16 * 4;
        float*       l  = s_tile + r * PITCH + c16 * 4;
        async_load_b128((void*)l, (void*)g);
    }
    wait_async0();        // this wave's ASYNC transfers have landed in LDS
    __syncthreads();      // make them visible to all waves in the workgroup

    // ---- Phase 2: transposed, fully coalesced, non-temporal writeback ----
    // out[b][d][n0+nl] = lds[nl][d]; tid-consecutive nl -> 256B contiguous
    // store segments. Output (128 MB) is write-once -> TH_NT keeps it from
    // evicting the L2-resident codebook.
    float* obase = out + (size_t)b * EMB_D * TOK_N + n0;
#pragma unroll 8
    for (int j = 0; j < 64; ++j) {
        const int flat = j * THREADS + tid;      // 0..16383
        const int nl   = flat & (TILE - 1);      // token within tile
        const int d    = flat >> 6;              // channel
        const float v  = s_tile[nl * PITCH + d];
#if __has_builtin(__builtin_nontemporal_store)
        __builtin_nontemporal_store(v, obase + (size_t)d * TOK_N + nl);
#else
        obase[(size_t)d * TOK_N + nl] = v;
#endif
    }
}

extern "C" void kernel_launch(void* const* d_in, const int* in_sizes, int n_in,
                              void* d_out, int out_size, void* d_ws, size_t ws_size,
                              hipStream_t stream) {
    const int*   idx = (const int*)d_in[0];     // (B,C,N) indices, flat
    const float* emb = (const float*)d_in[1];   // (K,D) codebook
    float*       out = (float*)d_out;           // (B,D,H,W) flat
    (void)d_ws; (void)ws_size; (void)n_in; (void)out_size;

    const int tokens = in_sizes[0];             // B*C*N = 131072
    dim3 grid(tokens / TILE), block(THREADS);
    hipLaunchKernelGGL(vq_embed_gather_tr_kernel, grid, block, 0, stream,
                       idx, emb, out);
}